// RESCLModel_33277406609792
// MI455X (gfx1250) — compile-verified
//
#include <hip/hip_runtime.h>
#include <hip/hip_bf16.h>

typedef __attribute__((ext_vector_type(2))) float v2f;
typedef __attribute__((ext_vector_type(8))) float v8f;

#define Bsz 512
#define Lsz 256
#define Hsz 768
#define TWOH 1536
#define NCLS 42
#define TAU 0.3f
#define LAM 0.5f

// ---------------- Kernel 1: masked max pool, float4 loads, masks in LDS -----
// grid = 512 (one batch row per block), block = 192 (each thread owns 4 h's)
__global__ void pool_kernel(const float* __restrict__ enc,
                            const float* __restrict__ m1,
                            const float* __restrict__ m2,
                            float* __restrict__ ent) {
    __shared__ float s1[Lsz];
    __shared__ float s2[Lsz];
    int b   = blockIdx.x;
    int tid = threadIdx.x;                 // 0..191
    for (int l = tid; l < Lsz; l += 192) {
        s1[l] = m1[(size_t)b * Lsz + l];
        s2[l] = m2[(size_t)b * Lsz + l];
    }
    __syncthreads();

    int h4 = tid * 4;
    const float4* ep = reinterpret_cast<const float4*>(enc + (size_t)b * Lsz * Hsz + h4);
    float4 mx1 = make_float4(-1000.f, -1000.f, -1000.f, -1000.f);
    float4 mx2 = mx1;
    #pragma unroll 4
    for (int l = 0; l < Lsz; ++l) {
        float4 x = ep[(size_t)l * (Hsz / 4)];
        bool a = (s1[l] != 0.0f);
        bool c = (s2[l] != 0.0f);
        mx1.x = fmaxf(mx1.x, a ? x.x : -1000.f);
        mx1.y = fmaxf(mx1.y, a ? x.y : -1000.f);
        mx1.z = fmaxf(mx1.z, a ? x.z : -1000.f);
        mx1.w = fmaxf(mx1.w, a ? x.w : -1000.f);
        mx2.x = fmaxf(mx2.x, c ? x.x : -1000.f);
        mx2.y = fmaxf(mx2.y, c ? x.y : -1000.f);
        mx2.z = fmaxf(mx2.z, c ? x.z : -1000.f);
        mx2.w = fmaxf(mx2.w, c ? x.w : -1000.f);
    }
    *reinterpret_cast<float4*>(ent + (size_t)b * TWOH + h4)       = mx1;
    *reinterpret_cast<float4*>(ent + (size_t)b * TWOH + Hsz + h4) = mx2;
}

// ---------------- Kernel 2: hdn = relu(ent @ W1 + b1), WMMA, 16x64/wave -----
__global__ void gemm1_kernel(const float* __restrict__ A,   // (512,1536)
                             const float* __restrict__ W,   // (1536,1536) K x N
                             const float* __restrict__ bias,
                             float* __restrict__ C) {       // (512,1536)
    int tn = blockIdx.x;          // 0..23 (64 cols each)
    int tm = blockIdx.y;          // 0..31
    int lane = threadIdx.x;       // wave32
    int half = lane >> 4;
    int l16  = lane & 15;
    int koff = half * 2;
    int m  = tm * 16 + l16;
    int n0 = tn * 64 + l16;

    v8f acc[4] = {};
    const float* arow = A + (size_t)m * TWOH;
    #pragma unroll 2
    for (int k = 0; k < TWOH; k += 4) {
        v2f a = *(const v2f*)(arow + k + koff);
        const float* w0 = W + (size_t)(k + koff) * TWOH + n0;
        #pragma unroll
        for (int q = 0; q < 4; ++q) {
            v2f bf;
            bf.x = w0[q * 16];
            bf.y = w0[TWOH + q * 16];
            acc[q] = __builtin_amdgcn_wmma_f32_16x16x4_f32(
                false, a, false, bf, (short)0, acc[q], false, false);
        }
    }
    #pragma unroll
    for (int q = 0; q < 4; ++q) {
        int n = n0 + q * 16;
        float bv = bias[n];
        #pragma unroll
        for (int r = 0; r < 8; ++r) {
            int mg = tm * 16 + half * 8 + r;
            C[(size_t)mg * TWOH + n] = fmaxf(acc[q][r] + bv, 0.0f);
        }
    }
}

// ---------------- Kernel 3: logits = hdn @ W2 + b2 (N=42 in 3 tiles/wave) ---
__global__ void gemm2_kernel(const float* __restrict__ A,   // (512,1536)
                             const float* __restrict__ W,   // (1536,42)
                             const float* __restrict__ bias,
                             float* __restrict__ out) {     // (512,42)
    int tm = blockIdx.x;          // 0..31
    int lane = threadIdx.x;
    int half = lane >> 4;
    int l16  = lane & 15;
    int koff = half * 2;
    int m = tm * 16 + l16;

    v8f acc[3] = {};
    const float* arow = A + (size_t)m * TWOH;
    #pragma unroll 2
    for (int k = 0; k < TWOH; k += 4) {
        v2f a = *(const v2f*)(arow + k + koff);
        const float* w0 = W + (size_t)(k + koff) * NCLS;
        #pragma unroll
        for (int q = 0; q < 3; ++q) {
            int n = q * 16 + l16;
            bool nok = (n < NCLS);
            v2f bf;
            bf.x = nok ? w0[n] : 0.0f;
            bf.y = nok ? w0[NCLS + n] : 0.0f;
            acc[q] = __builtin_amdgcn_wmma_f32_16x16x4_f32(
                false, a, false, bf, (short)0, acc[q], false, false);
        }
    }
    #pragma unroll
    for (int q = 0; q < 3; ++q) {
        int n = q * 16 + l16;
        if (n < NCLS) {
            float bv = bias[n];
            #pragma unroll
            for (int r = 0; r < 8; ++r) {
                int mg = tm * 16 + half * 8 + r;
                out[(size_t)mg * NCLS + n] = acc[q][r] + bv;
            }
        }
    }
}

// ---------------- Kernel 4: inverse row norms of ent ------------------------
__global__ void norm_kernel(const float* __restrict__ ent, float* __restrict__ invn) {
    __shared__ float sm[256];
    int b = blockIdx.x;
    int tid = threadIdx.x;
    const float* row = ent + (size_t)b * TWOH;
    float s = 0.0f;
    for (int k = tid; k < TWOH; k += 256) { float x = row[k]; s += x * x; }
    sm[tid] = s; __syncthreads();
    for (int off = 128; off > 0; off >>= 1) {
        if (tid < off) sm[tid] += sm[tid + off];
        __syncthreads();
    }
    if (tid == 0) invn[b] = 1.0f / sqrtf(sm[0]);
}

// ---------------- Kernel 5: cos = (ent @ ent^T)*invn_i*invn_j, 16x64/wave ---
__global__ void gram_kernel(const float* __restrict__ ent,
                            const float* __restrict__ invn,
                            float* __restrict__ cosm) {     // (512,512)
    int tn = blockIdx.x;          // 0..7 (64 cols each)
    int tm = blockIdx.y;          // 0..31
    int lane = threadIdx.x;
    int half = lane >> 4;
    int l16  = lane & 15;
    int koff = half * 2;
    int m  = tm * 16 + l16;
    int n0 = tn * 64 + l16;

    v8f acc[4] = {};
    const float* arow = ent + (size_t)m * TWOH;
    const float* brow0 = ent + (size_t)(n0 +  0) * TWOH;   // B[k][n] = ent[n][k]
    const float* brow1 = ent + (size_t)(n0 + 16) * TWOH;
    const float* brow2 = ent + (size_t)(n0 + 32) * TWOH;
    const float* brow3 = ent + (size_t)(n0 + 48) * TWOH;
    #pragma unroll 2
    for (int k = 0; k < TWOH; k += 4) {
        v2f a  = *(const v2f*)(arow  + k + koff);
        v2f b0 = *(const v2f*)(brow0 + k + koff);
        v2f b1 = *(const v2f*)(brow1 + k + koff);
        v2f b2 = *(const v2f*)(brow2 + k + koff);
        v2f b3 = *(const v2f*)(brow3 + k + koff);
        acc[0] = __builtin_amdgcn_wmma_f32_16x16x4_f32(false, a, false, b0, (short)0, acc[0], false, false);
        acc[1] = __builtin_amdgcn_wmma_f32_16x16x4_f32(false, a, false, b1, (short)0, acc[1], false, false);
        acc[2] = __builtin_amdgcn_wmma_f32_16x16x4_f32(false, a, false, b2, (short)0, acc[2], false, false);
        acc[3] = __builtin_amdgcn_wmma_f32_16x16x4_f32(false, a, false, b3, (short)0, acc[3], false, false);
    }
    #pragma unroll
    for (int q = 0; q < 4; ++q) {
        int n = n0 + q * 16;
        float inj = invn[n];
        #pragma unroll
        for (int r = 0; r < 8; ++r) {
            int mg = tm * 16 + half * 8 + r;
            cosm[(size_t)mg * Bsz + n] = acc[q][r] * invn[mg] * inj;
        }
    }
}

// ---------------- Kernel 6: denom[i], log(denom[i]), w[i] per row -----------
__global__ void denom_kernel(const float* __restrict__ cosm,
                             const int* __restrict__ labels,
                             float* __restrict__ logden,
                             float* __restrict__ wv) {
    __shared__ float sm[256];
    __shared__ float sc[256];
    int i = blockIdx.x;
    int tid = threadIdx.x;
    const float* row = cosm + (size_t)i * Bsz;
    int li = labels[i];
    float s = 0.0f, cnt = 0.0f;
    for (int j = tid; j < Bsz; j += 256) {
        s += __expf(row[j] / TAU);
        cnt += (labels[j] == li) ? 1.0f : 0.0f;
    }
    sm[tid] = s; sc[tid] = cnt; __syncthreads();
    for (int off = 128; off > 0; off >>= 1) {
        if (tid < off) { sm[tid] += sm[tid + off]; sc[tid] += sc[tid + off]; }
        __syncthreads();
    }
    if (tid == 0) {
        float denom = sm[0] - __expf(row[i] / TAU);
        logden[i] = logf(denom);
        float counts = sc[0];                      // includes self
        wv[i] = (counts > 1.0f) ? 1.0f / fmaxf(counts - 1.0f, 1.0f) : 0.0f;
    }
}

// ---------------- Kernel 7: per-row cross entropy ---------------------------
__global__ void ce_row_kernel(const float* __restrict__ logits,   // (512,42)
                              const int* __restrict__ labels,
                              float* __restrict__ cerow) {
    __shared__ float sm[64];
    int b = blockIdx.x;
    int tid = threadIdx.x;                        // 64 threads
    const float* row = logits + (size_t)b * NCLS;
    float v = (tid < NCLS) ? row[tid] : -3.402823e38f;
    sm[tid] = v; __syncthreads();
    for (int off = 32; off > 0; off >>= 1) {
        if (tid < off) sm[tid] = fmaxf(sm[tid], sm[tid + off]);
        __syncthreads();
    }
    float mx = sm[0]; __syncthreads();
    float e = (tid < NCLS) ? __expf(row[tid] - mx) : 0.0f;
    sm[tid] = e; __syncthreads();
    for (int off = 32; off > 0; off >>= 1) {
        if (tid < off) sm[tid] += sm[tid + off];
        __syncthreads();
    }
    if (tid == 0) {
        float lse = mx + logf(sm[0]);
        cerow[b] = -(row[labels[b]] - lse);
    }
}

// ---------------- Kernel 8: reduce ce rows -> mean scalar -------------------
__global__ void ce_reduce_kernel(const float* __restrict__ cerow, float* __restrict__ ce) {
    __shared__ float sm[256];
    int tid = threadIdx.x;
    float s = 0.0f;
    for (int b = tid; b < Bsz; b += 256) s += cerow[b];
    sm[tid] = s; __syncthreads();
    for (int off = 128; off > 0; off >>= 1) {
        if (tid < off) sm[tid] += sm[tid + off];
        __syncthreads();
    }
    if (tid == 0) ce[0] = sm[0] / (float)Bsz;
}

// ---------------- Kernel 9: final (B,B) loss matrix -------------------------
__global__ void loss_kernel(const float* __restrict__ cosm,
                            const float* __restrict__ logden,
                            const float* __restrict__ wv,
                            const int* __restrict__ labels,
                            const float* __restrict__ ce,
                            float* __restrict__ loss) {       // (512,512)
    int i = blockIdx.x;
    int tid = threadIdx.x;
    float cev = (1.0f - LAM) * ce[0];
    float wi = wv[i];
    int li = labels[i];
    const float* row = cosm + (size_t)i * Bsz;
    float* orow = loss + (size_t)i * Bsz;
    for (int j = tid; j < Bsz; j += 256) {
        float scl = 0.0f;
        if (j != i && labels[j] == li) {
            float lec = -(row[j] / TAU) + logden[j];   // -log(exp_cos/denom[j])
            scl = lec * wi;
        }
        orow[j] = cev + LAM * scl;
    }
}

extern "C" void kernel_launch(void* const* d_in, const int* in_sizes, int n_in,
                              void* d_out, int out_size, void* d_ws, size_t ws_size,
                              hipStream_t stream) {
    const float* enc    = (const float*)d_in[0];
    const float* m1     = (const float*)d_in[1];
    const float* m2     = (const float*)d_in[2];
    const int*   labels = (const int*)  d_in[3];
    const float* W1     = (const float*)d_in[4];
    const float* b1     = (const float*)d_in[5];
    const float* W2     = (const float*)d_in[6];
    const float* b2     = (const float*)d_in[7];

    float* out    = (float*)d_out;
    float* logits = out;                    // 512*42
    float* loss   = out + Bsz * NCLS;       // 512*512

    float* ws     = (float*)d_ws;
    float* ent    = ws;                                  // 512*1536
    float* hdn    = ent    + (size_t)Bsz * TWOH;         // 512*1536
    float* cosm   = hdn    + (size_t)Bsz * TWOH;         // 512*512
    float* invn   = cosm   + (size_t)Bsz * Bsz;          // 512
    float* logden = invn   + Bsz;                        // 512
    float* wv     = logden + Bsz;                        // 512
    float* cerow  = wv     + Bsz;                        // 512
    float* cesc   = cerow  + Bsz;                        // 1

    // 1. masked max pooling -> ent (float4 streaming, HBM-bound stage)
    pool_kernel<<<Bsz, 192, 0, stream>>>(enc, m1, m2, ent);
    // 2. hdn = relu(ent @ W1 + b1)   [WMMA, 16x64 per wave]
    gemm1_kernel<<<dim3(TWOH / 64, Bsz / 16), 32, 0, stream>>>(ent, W1, b1, hdn);
    // 3. logits = hdn @ W2 + b2      [WMMA, 3 guarded N tiles per wave]
    gemm2_kernel<<<Bsz / 16, 32, 0, stream>>>(hdn, W2, b2, logits);
    // 4. inverse row norms
    norm_kernel<<<Bsz, 256, 0, stream>>>(ent, invn);
    // 5. cosine similarity matrix    [WMMA, 16x64 per wave]
    gram_kernel<<<dim3(Bsz / 64, Bsz / 16), 32, 0, stream>>>(ent, invn, cosm);
    // 6. denom / weights per row
    denom_kernel<<<Bsz, 256, 0, stream>>>(cosm, labels, logden, wv);
    // 7. per-row CE
    ce_row_kernel<<<Bsz, 64, 0, stream>>>(logits, labels, cerow);
    // 8. CE mean
    ce_reduce_kernel<<<1, 256, 0, stream>>>(cerow, cesc);
    // 9. loss matrix
    loss_kernel<<<Bsz, 256, 0, stream>>>(cosm, logden, wv, labels, cesc, loss);
}